// PointCNN_4234837753921
// MI455X (gfx1250) — compile-verified
//
#include <hip/hip_runtime.h>
#include <cfloat>
#include <math.h>

typedef __attribute__((ext_vector_type(2))) float v2f;
typedef __attribute__((ext_vector_type(8))) float v8f;

__device__ __forceinline__ float eluf(float x) { return x > 0.f ? x : (__expf(x) - 1.f); }

// ---------------------------------------------------------------------------
// Generic fp32 WMMA GEMM:  C[b] = act(A[b](MxK) * B[b](KxN) + bias[b])
// One wave computes one 16x16 tile of C using v_wmma_f32_16x16x4_f32.
// GUARD=false: M%16==0, N%16==0, K%4==0 (K>=4), bias non-null -> raw pipelined
// loads, no exec masking.  GUARD=true: clamped addresses + value selects.
// ---------------------------------------------------------------------------
template <bool GUARD>
__global__ void wmma_gemm_kernel(const float* __restrict__ A, const float* __restrict__ Bm,
                                 const float* __restrict__ bias, float* __restrict__ C,
                                 int M, int N, int K,
                                 int lda, int ldb, int ldc,
                                 long bA, long bB, long bBias, long bC, int act)
{
    A  += (long)blockIdx.z * bA;
    Bm += (long)blockIdx.z * bB;
    C  += (long)blockIdx.z * bC;
    const float* bi = bias + (long)blockIdx.z * bBias;

    int mtiles = (M + 15) >> 4;
    int ntiles = (N + 15) >> 4;
    int wave = (int)(((long)blockIdx.x * blockDim.x + threadIdx.x) >> 5);
    if (wave >= mtiles * ntiles) return;          // wave-uniform exit
    int mt = wave / ntiles;
    int nt = wave - mt * ntiles;

    int lane = threadIdx.x & 31;
    int hi  = lane >> 4;
    int l16 = lane & 15;
    int arow = mt * 16 + l16;
    int col  = nt * 16 + l16;

    v8f acc = {};
    if (!GUARD) {
        const float* ap  = A + (long)arow * lda + 2 * hi;
        const float* bp0 = Bm + (long)(2 * hi) * ldb + col;
        v2f a, b, a1, b1;
        a.x = ap[0];  a.y = ap[1];
        b.x = bp0[0]; b.y = bp0[ldb];
        for (int kk = 4; kk < K; kk += 4) {
            const float* ap2 = ap + kk;
            const float* bp2 = bp0 + (long)kk * ldb;
            a1.x = ap2[0];  a1.y = ap2[1];
            b1.x = bp2[0];  b1.y = bp2[ldb];
            acc = __builtin_amdgcn_wmma_f32_16x16x4_f32(false, a, false, b, (short)0, acc,
                                                        false, false);
            a = a1; b = b1;
        }
        acc = __builtin_amdgcn_wmma_f32_16x16x4_f32(false, a, false, b, (short)0, acc,
                                                    false, false);
        float bv = bi[col];
#pragma unroll
        for (int r = 0; r < 8; ++r) {
            float v = acc[r] + bv;
            if (act == 1) v = eluf(v);
            C[(long)(mt * 16 + r + 8 * hi) * ldc + col] = v;
        }
    } else {
        int arC = arow < M ? arow : (M - 1);
        int coC = col  < N ? col  : (N - 1);
        bool av = arow < M, cv = col < N;
        for (int kk = 0; kk < K; kk += 4) {
            int ka = kk + 2 * hi;
            int k0 = ka     < K ? ka     : (K - 1);
            int k1 = ka + 1 < K ? ka + 1 : (K - 1);
            float a0 = A[(long)arC * lda + k0];
            float a1 = A[(long)arC * lda + k1];
            float b0 = Bm[(long)k0 * ldb + coC];
            float b1 = Bm[(long)k1 * ldb + coC];
            v2f a, b;
            a.x = (av && ka     < K) ? a0 : 0.f;
            a.y = (av && ka + 1 < K) ? a1 : 0.f;
            b.x = (cv && ka     < K) ? b0 : 0.f;
            b.y = (cv && ka + 1 < K) ? b1 : 0.f;
            acc = __builtin_amdgcn_wmma_f32_16x16x4_f32(false, a, false, b, (short)0, acc,
                                                        false, false);
        }
        if (cv) {
            float bv = bi[coC];
#pragma unroll
            for (int r = 0; r < 8; ++r) {
                int row = mt * 16 + r + 8 * hi;
                if (row < M) {
                    float v = acc[r] + bv;
                    if (act == 1) v = eluf(v);
                    C[(long)row * ldc + col] = v;
                }
            }
        }
    }
}

// ---------------------------------------------------------------------------
// Fused X-transform + depthwise conv per point:
//   xt(Cit x K) = [h | gather(x,nbr)](Cit x K) * t(K x K)    (WMMA)
//   o[c*dm+m]   = sum_k xt[c,k] * wd[c,m,k] + bd             (shfl_xor reduce)
// GUARD=false requires: K==16, cit%16==0, x != nullptr (conv3..conv6).
// ---------------------------------------------------------------------------
template <bool GUARD>
__global__ void xconv_apply_kernel(const float* __restrict__ h, const float* __restrict__ x,
                                   const int* __restrict__ nbr, const float* __restrict__ t,
                                   const float* __restrict__ wd, const float* __restrict__ bd,
                                   float* __restrict__ o, int P, int K, int Cd, int Cin, int dm)
{
    int cit = Cd + Cin;
    int mtiles = (cit + 15) >> 4;
    long wave = ((long)blockIdx.x * blockDim.x + threadIdx.x) >> 5;
    long total = 8L * P * mtiles;
    if (wave >= total) return;                    // wave-uniform

    int lane = threadIdx.x & 31;
    int hi  = lane >> 4;
    int l16 = lane & 15;
    int mt = (int)(wave % mtiles);
    long bp = wave / mtiles;
    int b = (int)(bp / P);

    const float* tp = t + bp * K * K;
    const int*   nb = nbr + bp * K;
    const float* hp = h + bp * K * Cd;
    int c0 = mt * 16;
    int am = c0 + l16;

    v8f acc = {};
    if (!GUARD) {
        // pointer-select gather: one v_cndmask on the address, unconditional load
        const float* hbase = hp + am;
        const float* xbase = x + (am - Cd);
        bool useH = am < Cd;
        long bP = (long)b * P;
        auto aptr = [&](int k) -> const float* {
            return useH ? (hbase + (long)k * Cd) : (xbase + (bP + nb[k]) * (long)Cin);
        };
        v2f a, b2, a1, b3;
        int ka = 2 * hi;
        a.x = *aptr(ka); a.y = *aptr(ka + 1);
        b2.x = tp[ka * 16 + l16]; b2.y = tp[(ka + 1) * 16 + l16];
#pragma unroll
        for (int kk = 4; kk < 16; kk += 4) {
            int kb = kk + 2 * hi;
            a1.x = *aptr(kb); a1.y = *aptr(kb + 1);
            b3.x = tp[kb * 16 + l16]; b3.y = tp[(kb + 1) * 16 + l16];
            acc = __builtin_amdgcn_wmma_f32_16x16x4_f32(false, a, false, b2, (short)0, acc,
                                                        false, false);
            a = a1; b2 = b3;
        }
        acc = __builtin_amdgcn_wmma_f32_16x16x4_f32(false, a, false, b2, (short)0, acc,
                                                    false, false);
        float* op = o + bp * (long)cit * dm;
#pragma unroll
        for (int r = 0; r < 8; ++r) {
            int cch = c0 + r + 8 * hi;
            for (int m = 0; m < dm; ++m) {
                float term = acc[r] * wd[((long)cch * dm + m) * 16 + l16];
                term += __shfl_xor(term, 1);
                term += __shfl_xor(term, 2);
                term += __shfl_xor(term, 4);
                term += __shfl_xor(term, 8);
                if (l16 == 0) op[cch * dm + m] = term + bd[cch * dm + m];
            }
        }
    } else {
        auto ldA = [&](int k) -> float {
            if (am >= cit || k >= K) return 0.f;
            if (am < Cd) return hp[k * Cd + am];
            return x[((long)b * P + nb[k]) * Cin + (am - Cd)];
        };
        for (int kk = 0; kk < K; kk += 4) {
            int ka = kk + 2 * hi;
            v2f a, b2;
            a.x = ldA(ka);
            a.y = ldA(ka + 1);
            b2.x = (ka     < K && l16 < K) ? tp[ka * K + l16]       : 0.f;
            b2.y = (ka + 1 < K && l16 < K) ? tp[(ka + 1) * K + l16] : 0.f;
            acc = __builtin_amdgcn_wmma_f32_16x16x4_f32(false, a, false, b2, (short)0, acc,
                                                        false, false);
        }
        float* op = o + bp * (long)cit * dm;
#pragma unroll
        for (int r = 0; r < 8; ++r) {
            int cch = c0 + r + 8 * hi;
            bool cv = cch < cit;
            for (int m = 0; m < dm; ++m) {
                float wv = (cv && l16 < K) ? wd[((long)cch * dm + m) * K + l16] : 0.f;
                float term = (l16 < K) ? acc[r] * wv : 0.f;
                term += __shfl_xor(term, 1);
                term += __shfl_xor(term, 2);
                term += __shfl_xor(term, 4);
                term += __shfl_xor(term, 8);
                if (cv && l16 == 0) op[cch * dm + m] = term + bd[cch * dm + m];
            }
        }
    }
}

// ---------------------------------------------------------------------------
// kNN with dilation: one block per query point; d2 table in LDS, Kd iterative
// argmin passes, keep every dil-th selection (self forced first, d2[q]=-1).
// ---------------------------------------------------------------------------
__global__ void knn_kernel(const float* __restrict__ pos, int* __restrict__ nbr,
                           int P, int K, int dil)
{
    __shared__ float d2[2048];
    __shared__ float rmn[256];
    __shared__ int   rmi[256];
    int b = blockIdx.y, q = blockIdx.x;
    const float* pb = pos + (long)b * P * 3;
    float qx = pb[q * 3 + 0], qy = pb[q * 3 + 1], qz = pb[q * 3 + 2];
    for (int j = threadIdx.x; j < P; j += blockDim.x) {
        float dx = pb[j * 3 + 0] - qx, dy = pb[j * 3 + 1] - qy, dz = pb[j * 3 + 2] - qz;
        d2[j] = (j == q) ? -1.f : (dx * dx + dy * dy + dz * dz);
    }
    __syncthreads();
    int Kd = K * dil;
    for (int s = 0; s < Kd; ++s) {
        float best = FLT_MAX; int bi = 0x7fffffff;
        for (int j = threadIdx.x; j < P; j += blockDim.x) {
            float d = d2[j];
            if (d < best) { best = d; bi = j; }
        }
        rmn[threadIdx.x] = best; rmi[threadIdx.x] = bi;
        __syncthreads();
        for (int st = 128; st > 0; st >>= 1) {
            if ((int)threadIdx.x < st) {
                float ov = rmn[threadIdx.x + st]; int oi = rmi[threadIdx.x + st];
                if (ov < rmn[threadIdx.x] ||
                    (ov == rmn[threadIdx.x] && oi < rmi[threadIdx.x])) {
                    rmn[threadIdx.x] = ov; rmi[threadIdx.x] = oi;
                }
            }
            __syncthreads();
        }
        if (threadIdx.x == 0) {
            int sel = rmi[0];
            if ((s % dil) == 0) nbr[((long)b * P + q) * K + s / dil] = sel;
            d2[sel] = FLT_MAX;
        }
        __syncthreads();
    }
}

// ---------------------------------------------------------------------------
// Deterministic FPS (start index 0): one block per graph, dmin table in LDS.
// ---------------------------------------------------------------------------
__global__ void fps_kernel(const float* __restrict__ pos, int* __restrict__ out, int P, int n)
{
    __shared__ float dmin[2048];
    __shared__ float rmx[256];
    __shared__ int   rxi[256];
    __shared__ int   lastS;
    int b = blockIdx.x;
    const float* pb = pos + (long)b * P * 3;
    for (int j = threadIdx.x; j < P; j += blockDim.x) dmin[j] = FLT_MAX;
    if (threadIdx.x == 0) lastS = 0;
    __syncthreads();
    for (int i = 0; i < n; ++i) {
        int last = lastS;
        if (threadIdx.x == 0) out[(long)b * n + i] = last;
        float lx = pb[last * 3 + 0], ly = pb[last * 3 + 1], lz = pb[last * 3 + 2];
        float best = -FLT_MAX; int bi = 0x7fffffff;
        for (int j = threadIdx.x; j < P; j += blockDim.x) {
            float dx = pb[j * 3 + 0] - lx, dy = pb[j * 3 + 1] - ly, dz = pb[j * 3 + 2] - lz;
            float d = fminf(dmin[j], dx * dx + dy * dy + dz * dz);
            dmin[j] = d;
            if (d > best) { best = d; bi = j; }
        }
        rmx[threadIdx.x] = best; rxi[threadIdx.x] = bi;
        __syncthreads();
        for (int st = 128; st > 0; st >>= 1) {
            if ((int)threadIdx.x < st) {
                float ov = rmx[threadIdx.x + st]; int oi = rxi[threadIdx.x + st];
                if (ov > rmx[threadIdx.x] ||
                    (ov == rmx[threadIdx.x] && oi < rxi[threadIdx.x])) {
                    rmx[threadIdx.x] = ov; rxi[threadIdx.x] = oi;
                }
            }
            __syncthreads();
        }
        if (threadIdx.x == 0) lastS = rxi[0];
        __syncthreads();
    }
}

__global__ void gather_kernel(const float* __restrict__ src, const int* __restrict__ idx,
                              float* __restrict__ dst, int Pout, int Pin, int C)
{
    long i = (long)blockIdx.x * blockDim.x + threadIdx.x;
    long total = 8L * Pout * C;
    if (i >= total) return;
    int c = (int)(i % C); long t = i / C; int p = (int)(t % Pout); int b = (int)(t / Pout);
    dst[i] = src[((long)b * Pin + idx[(long)b * Pout + p]) * C + c];
}

__global__ void rel_kernel(const float* __restrict__ pos, const int* __restrict__ nbr,
                           float* __restrict__ rel, int P, int K)
{
    long i = (long)blockIdx.x * blockDim.x + threadIdx.x;
    long total = 8L * P * K;
    if (i >= total) return;
    long bp = i / K; int p = (int)(bp % P); int b = (int)(bp / P);
    const float* pb = pos + (long)b * P * 3;
    int j = nbr[i];
    rel[i * 3 + 0] = pb[j * 3 + 0] - pb[p * 3 + 0];
    rel[i * 3 + 1] = pb[j * 3 + 1] - pb[p * 3 + 1];
    rel[i * 3 + 2] = pb[j * 3 + 2] - pb[p * 3 + 2];
}

__global__ void small_mm_kernel(const float* __restrict__ X, const float* __restrict__ W,
                                const float* __restrict__ bias, float* __restrict__ Y,
                                long rows, int cin, int cout, int act)
{
    long i = (long)blockIdx.x * blockDim.x + threadIdx.x;
    if (i >= rows * cout) return;
    int co = (int)(i % cout); long r = i / cout;
    float s = bias[co];
    const float* xr = X + r * cin;
    for (int ci = 0; ci < cin; ++ci) s += xr[ci] * W[ci * cout + co];
    if (act == 1) s = eluf(s);
    Y[i] = s;
}

// BatchNorm (training-mode, gamma=1,beta=0): two-pass sum/sumsq (C <= 256)
__global__ void bn_partial_kernel(const float* __restrict__ X, long N, int C,
                                  float* __restrict__ stat)
{
    __shared__ float ss[512];
    for (int c = threadIdx.x; c < 2 * C; c += blockDim.x) ss[c] = 0.f;
    __syncthreads();
    long total = N * (long)C;
    for (long i = (long)blockIdx.x * blockDim.x + threadIdx.x; i < total;
         i += (long)gridDim.x * blockDim.x) {
        int c = (int)(i % C);
        float v = X[i];
        atomicAdd(&ss[c], v);
        atomicAdd(&ss[C + c], v * v);
    }
    __syncthreads();
    for (int c = threadIdx.x; c < 2 * C; c += blockDim.x) atomicAdd(&stat[c], ss[c]);
}

__global__ void bn_apply_kernel(float* __restrict__ X, long N, int C,
                                const float* __restrict__ stat, int act)
{
    long i = (long)blockIdx.x * blockDim.x + threadIdx.x;
    if (i >= N * (long)C) return;
    int c = (int)(i % C);
    float mean = stat[c] / (float)N;
    float var  = stat[C + c] / (float)N - mean * mean;
    float v = (X[i] - mean) * rsqrtf(var + 1e-5f);
    if (act == 2) v = fmaxf(v, 0.f);
    X[i] = v;
}

__global__ void maxpool_kernel(const float* __restrict__ X, int ldx,
                               float* __restrict__ Y, int ldy, int P, int C)
{
    int i = blockIdx.x * blockDim.x + threadIdx.x;
    if (i >= 8 * C) return;
    int c = i % C, b = i / C;
    float m = -FLT_MAX;
    for (int p = 0; p < P; ++p) m = fmaxf(m, X[((long)b * P + p) * (long)ldx + c]);
    Y[(long)b * ldy + c] = m;
}

// Zero-padding copies (to keep every big GEMM on the unguarded WMMA path)
__global__ void pad_cols_kernel(const float* __restrict__ src, float* __restrict__ dst,
                                long rows, int cs, int cdst)
{
    long i = (long)blockIdx.x * blockDim.x + threadIdx.x;
    if (i >= rows * cdst) return;
    int c = (int)(i % cdst); long r = i / cdst;
    dst[i] = (c < cs) ? src[r * (long)cs + c] : 0.f;
}

__global__ void pad_rows_kernel(const float* __restrict__ src, float* __restrict__ dst,
                                long rs, long rdst, int cols)
{
    long i = (long)blockIdx.x * blockDim.x + threadIdx.x;
    if (i >= rdst * cols) return;
    long r = i / cols;
    dst[i] = (r < rs) ? src[i] : 0.f;
}

// ---------------------------------------------------------------------------
extern "C" void kernel_launch(void* const* d_in, const int* in_sizes, int n_in,
                              void* d_out, int out_size, void* d_ws, size_t ws_size,
                              hipStream_t stream)
{
    (void)in_sizes; (void)n_in; (void)out_size; (void)ws_size;
    const float* pos0 = (const float*)d_in[0];
    // params flattened as a jax pytree: per-conv dict keys sorted alphabetically
    enum { iB1 = 0, iB2, iBA, iBB, iBD, iBL, iBO, iW1, iW2, iWA, iWB, iWD, iWL, iWO };
    auto prm = [&](int conv, int which) -> const float* {
        return (const float*)d_in[2 + conv * 14 + which];
    };
    const float* lin_b = (const float*)d_in[2 + 6 * 14 + 0];
    const float* lin_w = (const float*)d_in[2 + 6 * 14 + 1];

    float* F = (float*)d_ws;
    auto take = [&](long n) -> float* { float* p = F; F += n; return p; };
    int*   nbr    = (int*)take(131072);
    int*   fidx   = (int*)take(6144);
    float* rel    = take(400000);
    float* h1     = take(2100000);
    float* h2     = take(2100000);
    float* tA     = take(1100000);
    float* tB     = take(1100000);
    float* obuf   = take(2100000);
    float* xA     = take(1450000);
    float* xB     = take(1450000);
    float* posB   = take(49152);
    float* posC   = take(49152);
    float* wopad  = take(680000);   // 960 x 704
    float* bopad  = take(704);
    float* lwpad  = take(180000);   // 692 x 256
    float* gbuf   = take(11072);    // 16 x 692
    float* tmpout = take(4096);     // 16 x 256
    float* stat   = take(1024);

    auto blocks = [](long n, int t) { return (unsigned)((n + t - 1) / t); };
    auto bn = [&](float* X, long N, int C, int act) {
        hipMemsetAsync(stat, 0, (size_t)(2 * C) * sizeof(float), stream);
        bn_partial_kernel<<<256, 256, 0, stream>>>(X, N, C, stat);
        bn_apply_kernel<<<blocks(N * (long)C, 256), 256, 0, stream>>>(X, N, C, stat, act);
    };
    auto gemm = [&](const float* A, const float* Bm, const float* bias, float* C,
                    int M, int N, int K, int lda, int ldb, int ldc,
                    long bA, long bB, long bBias, long bC, int batches, int act) {
        long waves = (long)((M + 15) / 16) * ((N + 15) / 16);
        unsigned gx = (unsigned)((waves * 32 + 255) / 256);
        if ((M % 16) == 0 && (N % 16) == 0 && (K % 4) == 0)
            wmma_gemm_kernel<false><<<dim3(gx, 1, batches), 256, 0, stream>>>(
                A, Bm, bias, C, M, N, K, lda, ldb, ldc, bA, bB, bBias, bC, act);
        else
            wmma_gemm_kernel<true><<<dim3(gx, 1, batches), 256, 0, stream>>>(
                A, Bm, bias, C, M, N, K, lda, ldb, ldc, bA, bB, bBias, bC, act);
    };

    // one-time zero-padded weight copies (deterministic, re-done every call)
    pad_cols_kernel<<<blocks(960L * 704, 256), 256, 0, stream>>>(prm(5, iWO), wopad, 960, 690, 704);
    pad_cols_kernel<<<blocks(704, 256), 256, 0, stream>>>(prm(5, iBO), bopad, 1, 690, 704);
    pad_rows_kernel<<<blocks(692L * 256, 256), 256, 0, stream>>>(lin_w, lwpad, 690, 692, 256);

    struct CC { int cin, cd, cout, K, dil; };
    const CC cc[6] = {{0, 8, 32, 8, 2},      {32, 8, 128, 12, 2},   {128, 32, 256, 16, 1},
                      {256, 64, 256, 16, 2}, {256, 64, 256, 16, 3}, {256, 64, 690, 16, 4}};
    const int Pv[6] = {2048, 768, 250, 250, 250, 250};

    const float* curpos = pos0;
    const float* curx = nullptr;
    float* outx = xA;
    const float* featfinal = xB;

    for (int c = 0; c < 6; ++c) {
        int P = Pv[c], K = cc[c].K, dil = cc[c].dil;
        int cd = cc[c].cd, cin = cc[c].cin, cout = cc[c].cout;
        int cit = cin + cd;
        int dm = (cout + cit - 1) / cit;
        long BP = 8L * P;
        long rows = BP * K;
        int KK = K * K;

        knn_kernel<<<dim3(P, 8), 256, 0, stream>>>(curpos, nbr, P, K, dil);
        rel_kernel<<<blocks(rows, 256), 256, 0, stream>>>(curpos, nbr, rel, P, K);

        // mlp1: (rel @ w1) elu bn ; (@ w2) elu bn
        small_mm_kernel<<<blocks(rows * cd, 256), 256, 0, stream>>>(
            rel, prm(c, iW1), prm(c, iB1), h1, rows, 3, cd, 1);
        bn(h1, rows, cd, 0);
        small_mm_kernel<<<blocks(rows * cd, 256), 256, 0, stream>>>(
            h1, prm(c, iW2), prm(c, iB2), h2, rows, cd, cd, 1);
        bn(h2, rows, cd, 0);

        // mlp2: t = bn(elu(rel @ wl)); grouped (wa,elu,bn); grouped (wb,bn)
        gemm(rel, prm(c, iWL), prm(c, iBL), tA, (int)BP, KK, 3 * K,
             3 * K, KK, KK, 0, 0, 0, 0, 1, 1);
        bn(tA, BP, KK, 0);
        gemm(tA, prm(c, iWA), prm(c, iBA), tB, (int)BP, K, K,
             KK, K, KK, (long)K, (long)KK, (long)K, (long)K, K, 1);
        bn(tB, BP, KK, 0);
        gemm(tB, prm(c, iWB), prm(c, iBB), tA, (int)BP, K, K,
             KK, K, KK, (long)K, (long)KK, (long)K, (long)K, K, 0);
        bn(tA, BP, KK, 0);

        // X-transform + depthwise (fused, WMMA per point), then output linear
        {
            int mtiles = (cit + 15) / 16;
            long waves = BP * mtiles;
            bool fastx = ((cit % 16) == 0) && (K == 16) && (curx != nullptr);
            if (fastx)
                xconv_apply_kernel<false><<<blocks(waves * 32, 256), 256, 0, stream>>>(
                    h2, curx, nbr, tA, prm(c, iWD), prm(c, iBD), obuf, P, K, cd, cin, dm);
            else
                xconv_apply_kernel<true><<<blocks(waves * 32, 256), 256, 0, stream>>>(
                    h2, curx, nbr, tA, prm(c, iWD), prm(c, iBD), obuf, P, K, cd, cin, dm);
        }
        {
            int Nout = (c == 5) ? 704 : cout;    // conv6 projection padded 690->704
            const float* Wo = (c == 5) ? wopad : prm(c, iWO);
            const float* Bo = (c == 5) ? bopad : prm(c, iBO);
            gemm(obuf, Wo, Bo, outx, (int)BP, Nout, cit * dm,
                 cit * dm, Nout, Nout, 0, 0, 0, 0, 1, 0);
        }
        if (c == 5) featfinal = outx;

        // FPS + gathers between resolutions; ping-pong feature buffers
        if (c == 0) {
            fps_kernel<<<8, 256, 0, stream>>>(curpos, fidx, 2048, 768);
            gather_kernel<<<blocks(8L * 768 * 3, 256), 256, 0, stream>>>(
                curpos, fidx, posB, 768, 2048, 3);
            gather_kernel<<<blocks(8L * 768 * 32, 256), 256, 0, stream>>>(
                outx, fidx, xB, 768, 2048, 32);
            curpos = posB; curx = xB; outx = xA;
        } else if (c == 1) {
            fps_kernel<<<8, 256, 0, stream>>>(curpos, fidx, 768, 250);
            gather_kernel<<<blocks(8L * 250 * 3, 256), 256, 0, stream>>>(
                curpos, fidx, posC, 250, 768, 3);
            gather_kernel<<<blocks(8L * 250 * 128, 256), 256, 0, stream>>>(
                outx, fidx, xB, 250, 768, 128);
            curpos = posC; curx = xB; outx = xA;
        } else if (c < 5) {
            const float* nx = outx;
            float* no = (outx == xA) ? xB : xA;
            curx = nx; outx = no;
        }
    }

    // global max pool (stride-704 input) -> padded final linear -> BN -> ReLU
    hipMemsetAsync(gbuf, 0, (size_t)16 * 692 * sizeof(float), stream);
    maxpool_kernel<<<blocks(8L * 690, 256), 256, 0, stream>>>(featfinal, 704, gbuf, 692, 250, 690);
    gemm(gbuf, lwpad, lin_b, tmpout, 16, 256, 692, 692, 256, 256, 0, 0, 0, 0, 1, 0);
    bn(tmpout, 8, 256, 2);
    hipMemcpyAsync(d_out, tmpout, (size_t)8 * 256 * sizeof(float),
                   hipMemcpyDeviceToDevice, stream);
}